// gnn_encoder_2302102471107
// MI455X (gfx1250) — compile-verified
//
#include <hip/hip_runtime.h>

typedef float v2f __attribute__((ext_vector_type(2)));
typedef float v8f __attribute__((ext_vector_type(8)));

#define P_N   20000
#define D_N   2000
#define SEQ_K 1024
#define HDIM  256
#define NE_PPI 400000
#define NE_DDI 160000
#define NE_DTI 120000

// ---------------------------------------------------------------------------
// GEMM: C[M,256] = A[M,K] @ W[K,256] (+bias) (+relu), fp32 via V_WMMA_F32_16X16X4_F32
// block = 128 threads (4 waves); wave w computes the 16x64 tile cols [64w,64w+64)
// grid.x = M/16 (M is always a multiple of 16 here).
// Fragment layouts per CDNA5 ISA 7.12.2:
//   A 16x4 : lane = m + 16*(k>>1), reg = k&1
//   B 4x16 : lane = n + 16*(k>>1), reg = k&1
//   C 16x16: reg r, lane L -> row = r + 8*(L>>4), col = L&15
// ---------------------------------------------------------------------------
__global__ void gemm_f32_wmma(const float* __restrict__ A,
                              const float* __restrict__ W,
                              const float* __restrict__ bias,
                              float* __restrict__ C,
                              int K, int do_relu) {
  const int lane  = threadIdx.x & 31;
  const int wave  = threadIdx.x >> 5;
  const int mbase = blockIdx.x << 4;
  const int n0    = wave << 6;
  const int mrow  = lane & 15;
  const int half  = lane >> 4;          // 0 or 1 (which k-pair / which row-half)

  v8f acc0 = {}, acc1 = {}, acc2 = {}, acc3 = {};
  const float* Arow = A + (size_t)(mbase + mrow) * K + 2 * half;

  for (int kb = 0; kb < K; kb += 4) {
    v2f a;
    a.x = Arow[kb];
    a.y = Arow[kb + 1];
    const float* Wr0 = W + (size_t)(kb + 2 * half) * HDIM + n0 + mrow;
    const float* Wr1 = Wr0 + HDIM;
    v2f b0, b1, b2, b3;
    b0.x = Wr0[0];  b0.y = Wr1[0];
    b1.x = Wr0[16]; b1.y = Wr1[16];
    b2.x = Wr0[32]; b2.y = Wr1[32];
    b3.x = Wr0[48]; b3.y = Wr1[48];
    acc0 = __builtin_amdgcn_wmma_f32_16x16x4_f32(false, a, false, b0, (short)0, acc0, false, false);
    acc1 = __builtin_amdgcn_wmma_f32_16x16x4_f32(false, a, false, b1, (short)0, acc1, false, false);
    acc2 = __builtin_amdgcn_wmma_f32_16x16x4_f32(false, a, false, b2, (short)0, acc2, false, false);
    acc3 = __builtin_amdgcn_wmma_f32_16x16x4_f32(false, a, false, b3, (short)0, acc3, false, false);
  }

  v8f accs[4] = {acc0, acc1, acc2, acc3};
  for (int t = 0; t < 4; ++t) {
    const int col = n0 + t * 16 + mrow;
    const float bv = bias ? bias[col] : 0.0f;
    #pragma unroll
    for (int r = 0; r < 8; ++r) {
      const int row = mbase + r + 8 * half;
      float v = accs[t][r] + bv;
      if (do_relu) v = fmaxf(v, 0.0f);
      C[(size_t)row * HDIM + col] = v;
    }
  }
}

// ---------------------------------------------------------------------------
// Graph helpers
// ---------------------------------------------------------------------------
__global__ void edge_count(const int* __restrict__ idx, int E, float* __restrict__ cnt) {
  int e = blockIdx.x * 256 + threadIdx.x;
  if (e < E) atomicAdd(&cnt[idx[e]], 1.0f);
}

__global__ void finalize_dinv(float* __restrict__ d, int N) {
  int i = blockIdx.x * 256 + threadIdx.x;
  if (i < N) d[i] = rsqrtf(d[i] + 1.0f);   // +1 self-loop; deg >= 1 always
}

// GCN neighbor aggregation (self-loop handled analytically in combine_ln).
// one block per edge, one thread per channel.
__global__ void scatter_gcn(const float* __restrict__ h,
                            const int* __restrict__ src,
                            const int* __restrict__ dst,
                            const float* __restrict__ dinv,
                            float* __restrict__ agg) {
  const int e = blockIdx.x;
  const int c = threadIdx.x;
  const int s = src[e], d = dst[e];
  const float w = dinv[s] * dinv[d];
  atomicAdd(&agg[(size_t)d * HDIM + c], h[(size_t)s * HDIM + c] * w);
}

// plain scatter-sum for scatter_mean (divide by count later)
__global__ void scatter_sum(const float* __restrict__ vals,
                            const int* __restrict__ from,
                            const int* __restrict__ to,
                            float* __restrict__ sum) {
  const int e = blockIdx.x;
  const int c = threadIdx.x;
  atomicAdd(&sum[(size_t)to[e] * HDIM + c], vals[(size_t)from[e] * HDIM + c]);
}

// ---------------------------------------------------------------------------
// Fused epilogue: t = agg + hW*dinv^2 (self loop) + gcn_b + lin (+ msum/max(cnt,1)) + 1e-6
// -> LayerNorm (eps 1e-5, optional affine) -> optional relu. one block per row.
// Safe for out == lin (each thread reads its own element before any write).
// ---------------------------------------------------------------------------
__global__ void combine_ln(const float* __restrict__ agg,
                           const float* __restrict__ hW,
                           const float* __restrict__ dinv,
                           const float* __restrict__ gcn_b,
                           const float* __restrict__ lin,
                           const float* __restrict__ msum,
                           const float* __restrict__ cnt,
                           const float* __restrict__ ln_g,
                           const float* __restrict__ ln_b,
                           float* __restrict__ out,
                           int do_relu) {
  __shared__ float red[HDIM];
  const int row = blockIdx.x;
  const int c = threadIdx.x;
  const size_t o = (size_t)row * HDIM + c;
  const float di = dinv[row];

  float t = agg[o] + hW[o] * di * di + gcn_b[c] + lin[o] + 1e-6f;
  if (msum) t += msum[o] / fmaxf(cnt[row], 1.0f);

  red[c] = t;
  __syncthreads();
  for (int s = HDIM / 2; s > 0; s >>= 1) {
    if (c < s) red[c] += red[c + s];
    __syncthreads();
  }
  const float mean = red[0] * (1.0f / HDIM);
  __syncthreads();

  const float dcen = t - mean;
  red[c] = dcen * dcen;
  __syncthreads();
  for (int s = HDIM / 2; s > 0; s >>= 1) {
    if (c < s) red[c] += red[c + s];
    __syncthreads();
  }
  const float var = red[0] * (1.0f / HDIM);

  float y = dcen * rsqrtf(var + 1e-5f);
  if (ln_g) y = y * ln_g[c] + ln_b[c];
  if (do_relu) y = fmaxf(y, 0.0f);
  out[o] = y;
}

// ---------------------------------------------------------------------------
// Host orchestration
// ---------------------------------------------------------------------------
extern "C" void kernel_launch(void* const* d_in, const int* in_sizes, int n_in,
                              void* d_out, int out_size, void* d_ws, size_t ws_size,
                              hipStream_t stream) {
  (void)in_sizes; (void)n_in; (void)out_size; (void)ws_size;

  auto F = [&](int i) { return (const float*)d_in[i]; };

  const float* seq      = F(0);
  const float* seq_init = F(1);
  const float* s1_gcn_W = F(2);  const float* s1_gcn_b = F(3);
  const float* s1_lin_W = F(4);  const float* s1_ln_g  = F(5);  const float* s1_ln_b = F(6);
  const float* s2_gcn_W = F(7);  const float* s2_gcn_b = F(8);
  const float* s2_lin_W = F(9);  const float* s2_ln_g  = F(10); const float* s2_ln_b = F(11);
  const float* drug_emb = F(12);
  const float* prot_emb = F(13);
  // pkg params: base + {pp_W,pp_b,dd_W,dd_b,td_W,td_b,pr_W,pr_b,dt_W,dt_b,dr_W,dr_b}
  const int P1 = 14, P2 = 26;
  const int* ppi = (const int*)d_in[38];
  const int* ddi = (const int*)d_in[39];
  const int* dti = (const int*)d_in[40];
  const int* ppi_src = ppi;            const int* ppi_dst = ppi + NE_PPI;
  const int* ddi_src = ddi;            const int* ddi_dst = ddi + NE_DDI;
  const int* dti_d   = dti;            const int* dti_p   = dti + NE_DTI;   // [drug, prot]

  const size_t PB = (size_t)P_N * HDIM;
  const size_t DB = (size_t)D_N * HDIM;

  float* w = (float*)d_ws;
  float* bufA  = w;            // [P,256]
  float* bufB  = bufA + PB;    // [P,256]
  float* bufC  = bufB + PB;    // [P,256]
  float* bufX  = bufC + PB;    // [P,256] current protein activation
  float* bufD1 = bufX + PB;    // [D,256]
  float* bufD2 = bufD1 + DB;   // [D,256]
  float* bufD3 = bufD2 + DB;   // [D,256]
  float* bufXD = bufD3 + DB;   // [D,256] current drug activation
  float* dinv_p = bufXD + DB;  // [P]
  float* dinv_d = dinv_p + P_N;// [D]
  float* cnt_d  = dinv_d + D_N;// [D]
  float* cnt_p  = cnt_d + D_N; // [P]

  float* out     = (float*)d_out;
  float* o_xseq  = out;
  float* o_prot  = out + PB;
  float* o_x1    = out + 2 * PB;
  float* o_x2    = out + 3 * PB;
  float* o_xdrug = out + 4 * PB;

  const int MP = P_N / 16, MD = D_N / 16;
  auto gemm = [&](const float* A, const float* Wm, const float* b, float* Cm,
                  int Mtiles, int K, int relu) {
    gemm_f32_wmma<<<Mtiles, 128, 0, stream>>>(A, Wm, b, Cm, K, relu);
  };
  auto zero = [&](float* p, size_t nfloats) {
    hipMemsetAsync(p, 0, nfloats * sizeof(float), stream);
  };

  // --- degrees & DTI counts (shared by all layers) ---
  zero(dinv_p, P_N); zero(dinv_d, D_N); zero(cnt_d, D_N); zero(cnt_p, P_N);
  edge_count<<<(NE_PPI + 255) / 256, 256, 0, stream>>>(ppi_dst, NE_PPI, dinv_p);
  edge_count<<<(NE_DDI + 255) / 256, 256, 0, stream>>>(ddi_dst, NE_DDI, dinv_d);
  edge_count<<<(NE_DTI + 255) / 256, 256, 0, stream>>>(dti_d,   NE_DTI, cnt_d);
  edge_count<<<(NE_DTI + 255) / 256, 256, 0, stream>>>(dti_p,   NE_DTI, cnt_p);
  finalize_dinv<<<(P_N + 255) / 256, 256, 0, stream>>>(dinv_p, P_N);
  finalize_dinv<<<(D_N + 255) / 256, 256, 0, stream>>>(dinv_d, D_N);

  // --- x_seq = relu(seq @ seq_init) (output 0, reused as s1 input) ---
  gemm(seq, seq_init, nullptr, o_xseq, MP, SEQ_K, 1);

  // --- seq layer 1 -> bufX (relu) ---
  gemm(o_xseq, s1_gcn_W, nullptr, bufA, MP, HDIM, 0);
  zero(bufB, PB);
  scatter_gcn<<<NE_PPI, 256, 0, stream>>>(bufA, ppi_src, ppi_dst, dinv_p, bufB);
  gemm(o_xseq, s1_lin_W, nullptr, bufC, MP, HDIM, 0);
  combine_ln<<<P_N, 256, 0, stream>>>(bufB, bufA, dinv_p, s1_gcn_b, bufC,
                                      nullptr, nullptr, s1_ln_g, s1_ln_b, bufX, 1);

  // --- seq layer 2 -> x1 (output 2, no relu) ---
  gemm(bufX, s2_gcn_W, nullptr, bufA, MP, HDIM, 0);
  zero(bufB, PB);
  scatter_gcn<<<NE_PPI, 256, 0, stream>>>(bufA, ppi_src, ppi_dst, dinv_p, bufB);
  gemm(bufX, s2_lin_W, nullptr, bufC, MP, HDIM, 0);
  combine_ln<<<P_N, 256, 0, stream>>>(bufB, bufA, dinv_p, s2_gcn_b, bufC,
                                      nullptr, nullptr, s2_ln_g, s2_ln_b, o_x1, 0);

  // --- x_prot_init (output 1) is a straight copy of prot_emb ---
  hipMemcpyAsync(o_prot, prot_emb, PB * sizeof(float), hipMemcpyDeviceToDevice, stream);

  // ===== pkg layer 1 (inputs drug_emb, prot_emb; both updates use OLD states) =====
  // drug update -> bufXD (relu, LN no affine)
  gemm(drug_emb, F(P1 + 2) /*dd_W*/, nullptr, bufD1, MD, HDIM, 0);
  zero(bufD2, DB);
  scatter_gcn<<<NE_DDI, 256, 0, stream>>>(bufD1, ddi_src, ddi_dst, dinv_d, bufD2);
  gemm(prot_emb, F(P1 + 8) /*dt_W*/, F(P1 + 9) /*dt_b*/, bufA, MP, HDIM, 0);
  zero(bufD3, DB);
  scatter_sum<<<NE_DTI, 256, 0, stream>>>(bufA, dti_p, dti_d, bufD3);
  gemm(drug_emb, F(P1 + 10) /*dr_W*/, F(P1 + 11) /*dr_b*/, bufXD, MD, HDIM, 0);
  combine_ln<<<D_N, 256, 0, stream>>>(bufD2, bufD1, dinv_d, F(P1 + 3) /*dd_b*/, bufXD,
                                      bufD3, cnt_d, nullptr, nullptr, bufXD, 1);

  // protein update -> bufX (relu, LN no affine)
  gemm(prot_emb, F(P1 + 0) /*pp_W*/, nullptr, bufA, MP, HDIM, 0);
  zero(bufB, PB);
  scatter_gcn<<<NE_PPI, 256, 0, stream>>>(bufA, ppi_src, ppi_dst, dinv_p, bufB);
  gemm(drug_emb, F(P1 + 4) /*td_W*/, F(P1 + 5) /*td_b*/, bufD1, MD, HDIM, 0);
  zero(bufC, PB);
  scatter_sum<<<NE_DTI, 256, 0, stream>>>(bufD1, dti_d, dti_p, bufC);
  gemm(prot_emb, F(P1 + 6) /*pr_W*/, F(P1 + 7) /*pr_b*/, bufX, MP, HDIM, 0);
  combine_ln<<<P_N, 256, 0, stream>>>(bufB, bufA, dinv_p, F(P1 + 1) /*pp_b*/, bufX,
                                      bufC, cnt_p, nullptr, nullptr, bufX, 1);

  // ===== pkg layer 2 (inputs bufXD, bufX) =====
  // drug update -> x_drug (output 4, no relu)
  gemm(bufXD, F(P2 + 2), nullptr, bufD1, MD, HDIM, 0);
  zero(bufD2, DB);
  scatter_gcn<<<NE_DDI, 256, 0, stream>>>(bufD1, ddi_src, ddi_dst, dinv_d, bufD2);
  gemm(bufX, F(P2 + 8), F(P2 + 9), bufA, MP, HDIM, 0);
  zero(bufD3, DB);
  scatter_sum<<<NE_DTI, 256, 0, stream>>>(bufA, dti_p, dti_d, bufD3);
  gemm(bufXD, F(P2 + 10), F(P2 + 11), o_xdrug, MD, HDIM, 0);
  combine_ln<<<D_N, 256, 0, stream>>>(bufD2, bufD1, dinv_d, F(P2 + 3), o_xdrug,
                                      bufD3, cnt_d, nullptr, nullptr, o_xdrug, 0);

  // protein update -> x2 (output 3, no relu)  [bufXD still holds OLD drug state]
  gemm(bufX, F(P2 + 0), nullptr, bufA, MP, HDIM, 0);
  zero(bufB, PB);
  scatter_gcn<<<NE_PPI, 256, 0, stream>>>(bufA, ppi_src, ppi_dst, dinv_p, bufB);
  gemm(bufXD, F(P2 + 4), F(P2 + 5), bufD1, MD, HDIM, 0);
  zero(bufC, PB);
  scatter_sum<<<NE_DTI, 256, 0, stream>>>(bufD1, dti_d, dti_p, bufC);
  gemm(bufX, F(P2 + 6), F(P2 + 7), o_x2, MP, HDIM, 0);
  combine_ln<<<P_N, 256, 0, stream>>>(bufB, bufA, dinv_p, F(P2 + 1), o_x2,
                                      bufC, cnt_p, nullptr, nullptr, o_x2, 0);
}